// Soft_F_Loss_65446711656629
// MI455X (gfx1250) — compile-verified
//
#include <hip/hip_runtime.h>
#include <hip/hip_bf16.h>
#include <math.h>

typedef __attribute__((ext_vector_type(2))) float v2f;
typedef __attribute__((ext_vector_type(8))) float v8f;

#define BB 4096   // batch
#define HH 512    // hidden
#define KK 32     // clusters
#define NPAIR ((KK*(KK-1))/2)   // 496

// ---------------------------------------------------------------------------
// Kernel 1: fused quad-WMMA GEMM via V_WMMA_F32_16X16X4_F32 (wave32).
//   M[h,k]  = sum_b hidden[b,h] * cluster[b,k]
//   S2[h,k] = sum_b hidden[b,h]^2 * cluster[b,k]^2
// One 16-row h-tile per 256-thread block; the SAME A loads feed both k-tiles
// (KK=32 -> k0=0 and k0=16), so each b-step issues 4 WMMAs on 6 loads.
// 8 waves each take a 512-deep b-slice; per-wave v8f accumulators are
// reduced across waves through LDS (4 tiles x 8 waves x 256 f32 = 32 KB).
// ---------------------------------------------------------------------------
__global__ __launch_bounds__(256) void gemm_sums_kernel(
    const float* __restrict__ hidden,   // [BB, HH] row-major
    const float* __restrict__ cluster,  // [BB, KK] row-major
    float* __restrict__ Mout,           // [HH, KK]
    float* __restrict__ S2out) {        // [HH, KK]
  __shared__ float ldsM0[8 * 256];
  __shared__ float ldsS0[8 * 256];
  __shared__ float ldsM1[8 * 256];
  __shared__ float ldsS1[8 * 256];

  const int tid  = threadIdx.x;
  const int wid  = tid >> 5;          // wave id (wave32)
  const int lane = tid & 31;
  const int half = lane >> 4;         // 0: K=0/1 , 1: K=2/3 (A/B layout)
  const int l    = lane & 15;

  const int h0 = blockIdx.x * 16;     // 32 h-tiles

  v8f accM0 = {}, accS0 = {};         // k = 0..15
  v8f accM1 = {}, accS1 = {};         // k = 16..31

  const int bBeg = wid * (BB / 8);
  const int bEnd = bBeg + (BB / 8);
  for (int b = bBeg; b < bEnd; b += 4) {
    const int br = b + 2 * half;
    // A tile (hidden^T): A[M=l][K=2*half(+1)] = hidden[(b+K)*HH + h0+l]
    v2f a;
    a.x = hidden[(size_t)br * HH + h0 + l];
    a.y = hidden[(size_t)(br + 1) * HH + h0 + l];
    // B tiles (cluster): B[K=2*half(+1)][N=l] = cluster[(b+K)*KK + k0+l]
    v2f b0, b1;
    b0.x = cluster[(size_t)br * KK + l];
    b0.y = cluster[(size_t)(br + 1) * KK + l];
    b1.x = cluster[(size_t)br * KK + 16 + l];
    b1.y = cluster[(size_t)(br + 1) * KK + 16 + l];

    accM0 = __builtin_amdgcn_wmma_f32_16x16x4_f32(
        false, a, false, b0, (short)0, accM0, false, false);
    accM1 = __builtin_amdgcn_wmma_f32_16x16x4_f32(
        false, a, false, b1, (short)0, accM1, false, false);

    const v2f a2  = a * a;
    const v2f b02 = b0 * b0;
    const v2f b12 = b1 * b1;
    accS0 = __builtin_amdgcn_wmma_f32_16x16x4_f32(
        false, a2, false, b02, (short)0, accS0, false, false);
    accS1 = __builtin_amdgcn_wmma_f32_16x16x4_f32(
        false, a2, false, b12, (short)0, accS1, false, false);
  }

#pragma unroll
  for (int v = 0; v < 8; ++v) {
    ldsM0[wid * 256 + v * 32 + lane] = accM0[v];
    ldsS0[wid * 256 + v * 32 + lane] = accS0[v];
    ldsM1[wid * 256 + v * 32 + lane] = accM1[v];
    ldsS1[wid * 256 + v * 32 + lane] = accS1[v];
  }
  __syncthreads();

  float sM0 = 0.f, sS0 = 0.f, sM1 = 0.f, sS1 = 0.f;
#pragma unroll
  for (int w = 0; w < 8; ++w) {
    sM0 += ldsM0[w * 256 + tid];
    sS0 += ldsS0[w * 256 + tid];
    sM1 += ldsM1[w * 256 + tid];
    sS1 += ldsS1[w * 256 + tid];
  }
  // D layout: D[v + 8*(lane>=16)][lane&15]
  const int v  = tid >> 5;
  const int ln = tid & 31;
  const int h  = h0 + v + 8 * (ln >> 4);
  const int k  = ln & 15;
  Mout[h * KK + k]       = sM0;
  S2out[h * KK + k]      = sS0;
  Mout[h * KK + 16 + k]  = sM1;
  S2out[h * KK + 16 + k] = sS1;
}

// ---------------------------------------------------------------------------
// Kernel 2: counts[k] = round(sum_b cluster[b,k])
// ---------------------------------------------------------------------------
__global__ __launch_bounds__(256) void counts_kernel(
    const float* __restrict__ cluster, float* __restrict__ counts) {
  __shared__ float red[256];
  const int k = blockIdx.x;
  float s = 0.f;
  for (int b = threadIdx.x; b < BB; b += 256) s += cluster[(size_t)b * KK + k];
  red[threadIdx.x] = s;
  __syncthreads();
  for (int st = 128; st > 0; st >>= 1) {
    if (threadIdx.x < st) red[threadIdx.x] += red[threadIdx.x + st];
    __syncthreads();
  }
  if (threadIdx.x == 0) counts[k] = roundf(red[0]);
}

// ---------------------------------------------------------------------------
// Regularized incomplete beta I_x(a,b), a = 0.5 (Lentz continued fraction).
// ---------------------------------------------------------------------------
__device__ float betacf_dev(float a, float b, float x) {
  const float FPMIN = 1e-30f;
  const float qab = a + b, qap = a + 1.f, qam = a - 1.f;
  float c = 1.f;
  float d = 1.f - qab * x / qap;
  if (fabsf(d) < FPMIN) d = FPMIN;
  d = 1.f / d;
  float h = d;
  for (int m = 1; m <= 200; ++m) {
    const float m2 = 2.f * m;
    float aa = m * (b - m) * x / ((qam + m2) * (a + m2));
    d = 1.f + aa * d; if (fabsf(d) < FPMIN) d = FPMIN;
    c = 1.f + aa / c; if (fabsf(c) < FPMIN) c = FPMIN;
    d = 1.f / d;
    h *= d * c;
    aa = -(a + m) * (qab + m) * x / ((a + m2) * (qap + m2));
    d = 1.f + aa * d; if (fabsf(d) < FPMIN) d = FPMIN;
    c = 1.f + aa / c; if (fabsf(c) < FPMIN) c = FPMIN;
    d = 1.f / d;
    const float del = d * c;
    h *= del;
    if (fabsf(del - 1.f) < 3e-7f) break;
  }
  return h;
}

__device__ float betainc_reg(float a, float b, float x) {
  if (x <= 0.f) return 0.f;
  if (x >= 1.f) return 1.f;
  const float lbt = lgammaf(a + b) - lgammaf(a) - lgammaf(b) +
                    a * logf(x) + b * log1pf(-x);
  const float bt = expf(lbt);
  if (x < (a + 1.f) / (a + b + 2.f))
    return bt * betacf_dev(a, b, x) / a;
  return 1.f - bt * betacf_dev(b, a, 1.f - x) / b;
}

// ---------------------------------------------------------------------------
// Kernel 3: one block per class pair (i<j). Compute xbetainc over H, take
// top-D via iterative LDS argmax, accumulate -sum(log(topD)) into out[0].
// ---------------------------------------------------------------------------
__global__ __launch_bounds__(256) void pair_kernel(
    const float* __restrict__ M, const float* __restrict__ S2,
    const float* __restrict__ counts, const int* __restrict__ dptr,
    float* __restrict__ out) {
  __shared__ float vals[HH];
  __shared__ float red[256];
  __shared__ int   redi[256];
  const int tid = threadIdx.x;

  // unrank blockIdx.x -> (i, j), i < j
  int i = 0, rem = blockIdx.x, row = KK - 1;
  while (rem >= row) { rem -= row; ++i; --row; }
  const int j = i + 1 + rem;

  const float ci = counts[i], cj = counts[j];
  const float pc = ci + cj;
  float d2 = pc - 2.0f;
  if (d2 == 0.0f) d2 += 1e-5f;
  const float bpar = 0.5f * d2;

  for (int h = tid; h < HH; h += 256) {
    const float mi = M[h * KK + i], mj = M[h * KK + j];
    const float wi = S2[h * KK + i] + (float)(BB - 2) * mi * mi;  // within_i
    const float wj = S2[h * KK + j] + (float)(BB - 2) * mj * mj;  // within_j
    const float diff = mi - mj;
    const float between = 0.25f * diff * diff * pc;
    float x = between / (between + wi + wj);
    x = fminf(fmaxf(x, 1e-37f), 1.0f - 1e-5f);
    vals[h] = betainc_reg(0.5f, bpar, x);
  }
  __syncthreads();

  int D = *dptr;
  if (D > HH) D = HH;
  float ssum = 0.f;  // meaningful in thread 0 only
  for (int t = 0; t < D; ++t) {
    const float v0 = vals[tid], v1 = vals[tid + 256];
    float bv; int bi;
    if (v1 > v0) { bv = v1; bi = tid + 256; } else { bv = v0; bi = tid; }
    red[tid] = bv; redi[tid] = bi;
    __syncthreads();
    for (int st = 128; st > 0; st >>= 1) {
      if (tid < st && red[tid + st] > red[tid]) {
        red[tid] = red[tid + st];
        redi[tid] = redi[tid + st];
      }
      __syncthreads();
    }
    if (tid == 0) {
      ssum += logf(red[0]);
      vals[redi[0]] = -INFINITY;
    }
    __syncthreads();
  }
  if (tid == 0) atomicAdd(out, -ssum);
}

__global__ void zero_kernel(float* out) { *out = 0.f; }

extern "C" void kernel_launch(void* const* d_in, const int* in_sizes, int n_in,
                              void* d_out, int out_size, void* d_ws, size_t ws_size,
                              hipStream_t stream) {
  const float* hidden  = (const float*)d_in[0];  // [4096, 512] f32
  const float* cluster = (const float*)d_in[1];  // [4096, 32]  f32
  const int*   dptr    = (const int*)d_in[2];    // scalar d = 20
  float* out = (float*)d_out;

  float* Mbuf  = (float*)d_ws;           // [HH*KK]
  float* S2buf = Mbuf + HH * KK;         // [HH*KK]
  float* cnts  = S2buf + HH * KK;        // [KK]

  zero_kernel<<<1, 1, 0, stream>>>(out);
  gemm_sums_kernel<<<HH / 16, 256, 0, stream>>>(hidden, cluster, Mbuf, S2buf);
  counts_kernel<<<KK, 256, 0, stream>>>(cluster, cnts);
  pair_kernel<<<NPAIR, 256, 0, stream>>>(Mbuf, S2buf, cnts, dptr, out);
}